// chamferDist_68685116998012
// MI455X (gfx1250) — compile-verified
//
#include <hip/hip_runtime.h>

typedef __attribute__((ext_vector_type(2))) float v2f;
typedef __attribute__((ext_vector_type(8))) float v8f;

#define WAVES         8                      // waves per block
#define ROWS_PER_WAVE 32                     // two 16-row A tiles per wave
#define TILE_ROWS     (WAVES * ROWS_PER_WAVE)  // 256 rows of X per block
#define CHUNK         256                    // Y points staged in LDS per iter

// r = min(r, a, b) in one VALU op; avoids the compiler's NaN-canonicalize
// (v_max x,x,x) bloat around llvm.minnum on WMMA results.
__device__ __forceinline__ void min3_acc(float& r, float a, float b) {
    asm("v_min3_num_f32 %0, %0, %1, %2" : "+v"(r) : "v"(a), "v"(b));
}

// For each point x in X (per batch), find min_y ||x - y||^2 over all Y.
// V_WMMA_F32_16X16X4_F32 homogeneous trick:
//   A' = [-2x, -2y, -2z, 1] (16x4), B' = [bx, by, bz, |b|^2] (4x16), C = |a|^2
//   => D = A'B' + C = |a|^2 + |b|^2 - 2 a.b   (full 16x16 tile per WMMA)
// LDS chunk stored as float4 (x, y, z, |b|^2): each lane's B fragment is a
// direct ds_load_b64 (low lanes .xy = K0,1; high lanes .zw = K2,3).
__global__ __launch_bounds__(256) void chamfer_nn_min_wmma(
    const float* __restrict__ X, const float* __restrict__ Y,
    float* __restrict__ out, int nx, int ny)
{
    __shared__ float4 sY[CHUNK];

    const int tid  = threadIdx.x;
    const int lane = tid & 31;              // wave32
    const int wave = tid >> 5;
    const int hi   = (lane >= 16) ? 1 : 0;  // high lane-half: K in {2,3}
    const int l15  = lane & 15;
    const int batch = blockIdx.y;
    const int row_base = blockIdx.x * TILE_ROWS + wave * ROWS_PER_WAVE;

    const float* Xb = X + (size_t)batch * nx * 3;
    const float* Yb = Y + (size_t)batch * ny * 3;

    // ---- Build two A' tiles per wave (rows row_base..+15, +16..+31).
    // ISA 16x4 f32 A layout: lane L, VGPR v holds row M=(L&15), K = 2*hi + v.
    v2f Areg[2];
    v8f cinit[2];
    float rmin0[8], rmin1[8];
#pragma unroll
    for (int s = 0; s < 2; ++s) {
        const float* xp = Xb + (size_t)(row_base + s * 16 + l15) * 3;
        const float ax = xp[0], ay = xp[1], az = xp[2];
        const float aa = ax * ax + ay * ay + az * az;
        Areg[s][0] = hi ? (-2.0f * az) : (-2.0f * ax);  // K=2 : K=0
        Areg[s][1] = hi ? 1.0f         : (-2.0f * ay);  // K=3 : K=1
        // C[vgpr i][lane] = |a|^2 of row (i + 8*hi); broadcast via shfl.
#pragma unroll
        for (int i = 0; i < 8; ++i)
            cinit[s][i] = __shfl(aa, i + (hi ? 8 : 0), 32);
    }
#pragma unroll
    for (int i = 0; i < 8; ++i) { rmin0[i] = 3.4e38f; rmin1[i] = 3.4e38f; }

    const float2* sB = reinterpret_cast<const float2*>(sY);

    // ---- Sweep all of Y in LDS-staged chunks of 256 points.
    for (int base = 0; base < ny; base += CHUNK) {
        {
            const float* yp = Yb + (size_t)(base + tid) * 3;
            const float bx = yp[0], by = yp[1], bz = yp[2];
            sY[tid] = make_float4(bx, by, bz, bx * bx + by * by + bz * bz);
        }
        __syncthreads();

#pragma unroll 2
        for (int t = 0; t < CHUNK; t += 32) {
            // Two B fragments per iteration (ds_load_2addr_b64 candidates).
            const float2 pa = sB[2 * (t + l15) + hi];
            const float2 pb = sB[2 * (t + 16 + l15) + hi];
            v2f Ba; Ba[0] = pa.x; Ba[1] = pa.y;
            v2f Bb; Bb[0] = pb.x; Bb[1] = pb.y;

            // Four independent WMMAs: {2 A tiles} x {2 B fragments}.
            v8f d0a = __builtin_amdgcn_wmma_f32_16x16x4_f32(
                false, Areg[0], false, Ba, (short)0, cinit[0], false, false);
            v8f d1a = __builtin_amdgcn_wmma_f32_16x16x4_f32(
                false, Areg[1], false, Ba, (short)0, cinit[1], false, false);
            v8f d0b = __builtin_amdgcn_wmma_f32_16x16x4_f32(
                false, Areg[0], false, Bb, (short)0, cinit[0], false, false);
            v8f d1b = __builtin_amdgcn_wmma_f32_16x16x4_f32(
                false, Areg[1], false, Bb, (short)0, cinit[1], false, false);

#pragma unroll
            for (int i = 0; i < 8; ++i) min3_acc(rmin0[i], d0a[i], d0b[i]);
#pragma unroll
            for (int i = 0; i < 8; ++i) min3_acc(rmin1[i], d1a[i], d1b[i]);
        }
        __syncthreads();
    }

    // ---- Min-reduce across the 16 lanes of each half (row m of a tile lives
    // in VGPR (m&7) across one lane-half). XOR masks 1..8 stay within halves.
#pragma unroll
    for (int m = 1; m <= 8; m <<= 1) {
#pragma unroll
        for (int i = 0; i < 8; ++i) {
            rmin0[i] = fminf(rmin0[i], __shfl_xor(rmin0[i], m, 32));
            rmin1[i] = fminf(rmin1[i], __shfl_xor(rmin1[i], m, 32));
        }
    }
    // lane 0 -> rows +0..7 ; lane 16 -> rows +8..15 of each tile.
    if (l15 == 0) {
        float* ob0 = out + (size_t)batch * nx + row_base + hi * 8;
        float* ob1 = ob0 + 16;
#pragma unroll
        for (int i = 0; i < 8; ++i) {
            ob0[i] = fmaxf(rmin0[i], 0.0f);   // clamp fp noise from the
            ob1[i] = fmaxf(rmin1[i], 0.0f);   // aa+bb-2ab expansion
        }
    }
}

extern "C" void kernel_launch(void* const* d_in, const int* in_sizes, int n_in,
                              void* d_out, int out_size, void* d_ws, size_t ws_size,
                              hipStream_t stream) {
    (void)n_in; (void)d_ws; (void)ws_size; (void)out_size;
    const float* xyz1 = (const float*)d_in[0];   // [B, N, 3] f32
    const float* xyz2 = (const float*)d_in[1];   // [B, M, 3] f32
    float* out = (float*)d_out;                  // dist1 [B,N] ++ dist2 [B,M]

    const int B = 8;
    const int N = in_sizes[0] / (3 * B);
    const int M = in_sizes[1] / (3 * B);

    dim3 block(256);
    // dist1: for each x in xyz1, nearest in xyz2
    dim3 grid1(N / TILE_ROWS, B);
    chamfer_nn_min_wmma<<<grid1, block, 0, stream>>>(xyz1, xyz2, out, N, M);
    // dist2: for each y in xyz2, nearest in xyz1
    dim3 grid2(M / TILE_ROWS, B);
    chamfer_nn_min_wmma<<<grid2, block, 0, stream>>>(xyz2, xyz1,
                                                     out + (size_t)B * N, M, N);
}